// ConditionalDecoder_40690520162971
// MI455X (gfx1250) — compile-verified
//
#include <hip/hip_runtime.h>
#include <hip/hip_bf16.h>
#include <math.h>

#define BATCH   128
#define SEQ     64
#define DEC     512
#define ENC2    1024
#define EMBD    256
#define VOCAB   32000
#define ATTN    256
#define NSTATES 30
#define MAXLEN  30
#define BOS_TOK 1
#define EOS_TOK 2

typedef float v2f __attribute__((ext_vector_type(2)));
typedef float v8f __attribute__((ext_vector_type(8)));

// ---------------------------------------------------------------------------
// fp32 GEMM on the CDNA5 tensor pipe: V_WMMA_F32_16X16X4_F32.
//   C[m,n] = sum_k A[m,k] * B(k,n)  (+ bias[n])
// BMODE 0: weight stored [N,K] (torch nn.Linear; computes x @ W.T)
// BMODE 1: weight stored [K,N] (computes x @ W)
//
// grid (N/64, M/128), block = 256 (8 waves). Each block computes a 128x64 C
// tile; wave w owns rows 16w..16w+15. The 64x16 B chunk is staged into LDS
// once per block (float4 coalesced loads + next-chunk global_prefetch) and
// reused by all 8 waves; fragments are read back as ds_load_b64 with a
// 20-float row pitch so the 16-lane reads land in 16 distinct banks.
//
// A frag (16x4 f32):  lane l, reg j -> A[M = l&15][K = 2*(l>>4)+j]
// B frag (4x16 f32):  lane l, reg j -> B[K = 2*(l>>4)+j][N = l&15]
// C frag (16x16 f32): lane l, reg r -> C[M = r + 8*(l>>4)][N = l&15]
// ---------------------------------------------------------------------------
#define BPITCH 20   // LDS row pitch (floats): 16B-aligned, conflict-free

template <int BMODE>
__global__ __launch_bounds__(256)
void gemm_wmma_f32(const float* __restrict__ A, const float* __restrict__ W,
                   const float* __restrict__ bias, float* __restrict__ C,
                   int M, int N, int K, int ldc)
{
  __shared__ float lds_b[64][BPITCH];   // [n within tile][k within chunk]

  const int tid  = threadIdx.x;
  const int lane = tid & 31;
  const int half = lane >> 4;          // 0 or 1
  const int mn   = lane & 15;
  const int tn   = blockIdx.x * 64;
  const int tm   = blockIdx.y * 128 + (tid >> 5) * 16;   // wave's 16 rows

  v8f acc[4];
  acc[0] = (v8f){}; acc[1] = (v8f){}; acc[2] = (v8f){}; acc[3] = (v8f){};

  const float* __restrict__ arow = A + (size_t)(tm + mn) * K;

  for (int k0 = 0; k0 < K; k0 += 16) {
    __syncthreads();
    // ---- stage 64(cols) x 16(k) chunk of B into LDS, cooperatively ----
    if (BMODE == 0) {
      // W is [N,K]: thread t loads 4 consecutive k of row (tn + t/4)
      const int n  = tid >> 2;
      const int kc = (tid & 3) * 4;
      const float4 v = *(const float4*)(W + (size_t)(tn + n) * K + k0 + kc);
      *(float4*)&lds_b[n][kc] = v;
      if (k0 + 16 < K)
        __builtin_prefetch(W + (size_t)(tn + n) * K + k0 + 16 + kc, 0, 3);
    } else {
      // W is [K,N]: thread t loads 4 consecutive n of row (k0 + t/16)
      const int k  = tid >> 4;
      const int n4 = (tid & 15) * 4;
      const float4 v = *(const float4*)(W + (size_t)(k0 + k) * N + tn + n4);
      lds_b[n4 + 0][k] = v.x;
      lds_b[n4 + 1][k] = v.y;
      lds_b[n4 + 2][k] = v.z;
      lds_b[n4 + 3][k] = v.w;
      if (k0 + 16 < K)
        __builtin_prefetch(W + (size_t)(k0 + 16 + k) * N + tn + n4, 0, 3);
    }
    __syncthreads();

    // ---- 16 WMMAs per chunk per wave (4 k-steps x 4 column sub-tiles) ----
#pragma unroll
    for (int kk = 0; kk < 16; kk += 4) {
      const int ka = kk + half * 2;
      v2f a;
      a.x = arow[k0 + ka];
      a.y = arow[k0 + ka + 1];
#pragma unroll
      for (int j = 0; j < 4; ++j) {
        v2f b;
        b.x = lds_b[16 * j + mn][ka];
        b.y = lds_b[16 * j + mn][ka + 1];
        acc[j] = __builtin_amdgcn_wmma_f32_16x16x4_f32(
            false, a, false, b, (short)0, acc[j], false, false);
      }
    }
  }

#pragma unroll
  for (int j = 0; j < 4; ++j) {
    const int ncol = tn + 16 * j + mn;
    const float bv = bias ? bias[ncol] : 0.0f;
#pragma unroll
    for (int r = 0; r < 8; ++r) {
      const int m = tm + r + 8 * half;
      C[(size_t)m * ldc + ncol] = acc[j][r] + bv;
    }
  }
}

// ---------------------------------------------------------------------------
// Per-call state init: stopped=0, finals=-1, inp=BOS
// ---------------------------------------------------------------------------
__global__ void init_state_kernel(int* __restrict__ inp, int* __restrict__ stopped,
                                  int* __restrict__ finals)
{
  int b = threadIdx.x;
  if (b < BATCH) { inp[b] = BOS_TOK; stopped[b] = 0; finals[b] = -1; }
}

// ---------------------------------------------------------------------------
// Embedding gather: emb[b, :] = table[inp[b], :]
// ---------------------------------------------------------------------------
__global__ void embed_kernel(const float* __restrict__ table,
                             const int* __restrict__ inp,
                             float* __restrict__ emb)
{
  int b = blockIdx.x;
  int e = threadIdx.x;              // 256 = EMBD
  emb[(size_t)b * EMBD + e] = table[(size_t)inp[b] * EMBD + e];
}

// ---------------------------------------------------------------------------
// GRU gate fusion (biases already folded into gi/gh by the GEMMs):
//   r = sig(gi_r+gh_r); z = sig(gi_z+gh_z); n = tanh(gi_n + r*gh_n)
//   out = (1-z)*n + z*h   [optional relu for the top cell feeding the fc]
// ---------------------------------------------------------------------------
__global__ void gru_combine_kernel(const float* __restrict__ gi,
                                   const float* __restrict__ gh,
                                   const float* __restrict__ h,
                                   float* __restrict__ out, int H, int do_relu)
{
  int idx = blockIdx.x * blockDim.x + threadIdx.x;
  if (idx >= BATCH * H) return;
  int b = idx / H, j = idx - b * H;
  const float* gib = gi + (size_t)b * 3 * H;
  const float* ghb = gh + (size_t)b * 3 * H;
  float r = 1.0f / (1.0f + expf(-(gib[j] + ghb[j])));
  float z = 1.0f / (1.0f + expf(-(gib[H + j] + ghb[H + j])));
  float n = tanhf(gib[2 * H + j] + r * ghb[2 * H + j]);
  float o = (1.0f - z) * n + z * h[idx];
  if (do_relu) o = fmaxf(o, 0.0f);
  out[idx] = o;
}

// ---------------------------------------------------------------------------
// Fused attention per batch row (grid=BATCH, block=256):
//   scores[s] = sum_a tanh(enc_proj[b,s,a] + dec_proj[b,a]) * v_a[a]
//   w = softmax(scores); write w to attn_weights output slice
//   attn_vec[b,e] = sum_s w[s] * enc_states[b,s,e]
// ---------------------------------------------------------------------------
__global__ void attention_kernel(const float* __restrict__ enc_proj,   // [B,S,ATTN]
                                 const float* __restrict__ dec_proj,   // [B,ATTN]
                                 const float* __restrict__ v_a,        // [ATTN]
                                 const float* __restrict__ enc_states, // [B,S,ENC2]
                                 float* __restrict__ w_out,            // + b*w_stride
                                 int w_stride,
                                 float* __restrict__ attn_vec)         // [B,ENC2]
{
  const int b = blockIdx.x;
  const int tid = threadIdx.x;      // 256
  __shared__ float partial[256];
  __shared__ float sc[SEQ];
  __shared__ float inv_sum;

  // partial score: 4 chunks of 64 attn-dims per (s)
  const int s = tid & 63;
  const int chunk = tid >> 6;
  const float* ep = enc_proj + ((size_t)b * SEQ + s) * ATTN;
  const float* dp = dec_proj + (size_t)b * ATTN;
  float acc = 0.0f;
  for (int a = chunk * 64; a < chunk * 64 + 64; ++a)
    acc += tanhf(ep[a] + dp[a]) * v_a[a];
  partial[tid] = acc;
  __syncthreads();
  if (tid < SEQ)
    sc[tid] = partial[tid] + partial[tid + 64] + partial[tid + 128] + partial[tid + 192];
  __syncthreads();
  if (tid == 0) {
    float mx = sc[0];
    for (int i = 1; i < SEQ; ++i) mx = fmaxf(mx, sc[i]);
    float sm = 0.0f;
    for (int i = 0; i < SEQ; ++i) { sc[i] = expf(sc[i] - mx); sm += sc[i]; }
    inv_sum = 1.0f / sm;
  }
  __syncthreads();
  if (tid < SEQ) {
    float w = sc[tid] * inv_sum;
    sc[tid] = w;
    w_out[(size_t)b * w_stride + tid] = w;
  }
  __syncthreads();
  // context vector: each thread produces 4 of the 1024 outputs
#pragma unroll
  for (int j = 0; j < 4; ++j) {
    const int e = tid + j * 256;
    const float* es = enc_states + (size_t)b * SEQ * ENC2 + e;
    float cv = 0.0f;
    for (int ss = 0; ss < SEQ; ++ss) cv += sc[ss] * es[(size_t)ss * ENC2];
    attn_vec[(size_t)b * ENC2 + e] = cv;
  }
}

// ---------------------------------------------------------------------------
// Greedy argmax over VOCAB (first-index tie break, matching jnp.argmax),
// plus stopped/finals bookkeeping with scan index t.
// grid = BATCH, block = 256. logits row b at logits + b*ld.
// ---------------------------------------------------------------------------
__global__ void argmax_update_kernel(const float* __restrict__ logits, int ld,
                                     int* __restrict__ inp, int* __restrict__ stopped,
                                     int* __restrict__ finals, int t)
{
  const int b = blockIdx.x;
  const int tid = threadIdx.x;
  const float* row = logits + (size_t)b * ld;
  float best = -INFINITY;
  int bi = VOCAB;
  for (int v = tid; v < VOCAB; v += 256) {
    float x = row[v];
    if (x > best) { best = x; bi = v; }   // strictly greater -> first index wins
  }
  __shared__ float bv[256];
  __shared__ int bix[256];
  bv[tid] = best; bix[tid] = bi;
  __syncthreads();
  for (int off = 128; off > 0; off >>= 1) {
    if (tid < off) {
      if (bv[tid + off] > bv[tid] ||
          (bv[tid + off] == bv[tid] && bix[tid + off] < bix[tid])) {
        bv[tid] = bv[tid + off]; bix[tid] = bix[tid + off];
      }
    }
    __syncthreads();
  }
  if (tid == 0) {
    int idx = bix[0];
    inp[b] = idx;
    int st = stopped[b] | (idx == EOS_TOK);
    stopped[b] = st;
    if (st && finals[b] == -1) finals[b] = t;
  }
}

// final_inds = finals==-1 ? MAXLEN : finals   (written as f32 into the flat out)
__global__ void finalize_kernel(const int* __restrict__ finals, float* __restrict__ out)
{
  int b = threadIdx.x;
  if (b < BATCH) {
    int f = finals[b];
    out[b] = (float)(f == -1 ? MAXLEN : f);
  }
}

// ---------------------------------------------------------------------------
extern "C" void kernel_launch(void* const* d_in, const int* in_sizes, int n_in,
                              void* d_out, int out_size, void* d_ws, size_t ws_size,
                              hipStream_t stream)
{
  (void)in_sizes; (void)n_in; (void)out_size; (void)ws_size;

  const float* hidden     = (const float*)d_in[0];   // [B,DEC]
  const float* enc_states = (const float*)d_in[1];   // [B,S,ENC2]
  const float* emb_table  = (const float*)d_in[2];   // [VOCAB,EMBD]
  const float* Wih_b      = (const float*)d_in[3];   // [3*DEC,EMBD]
  const float* Whh_b      = (const float*)d_in[4];   // [3*DEC,DEC]
  const float* bih_b      = (const float*)d_in[5];
  const float* bhh_b      = (const float*)d_in[6];
  const float* W_enc_a    = (const float*)d_in[7];   // [ENC2,ATTN]
  const float* W_dec_a    = (const float*)d_in[8];   // [DEC,ATTN]
  const float* v_a        = (const float*)d_in[9];   // [ATTN]
  const float* Wih_t      = (const float*)d_in[10];  // [3*ENC2,DEC]
  const float* Whh_t      = (const float*)d_in[11];  // [3*ENC2,ENC2]
  const float* bih_t      = (const float*)d_in[12];
  const float* bhh_t      = (const float*)d_in[13];
  const float* W_fc       = (const float*)d_in[14];  // [DEC,ENC2]
  const float* b_fc       = (const float*)d_in[15];
  const float* W_out      = (const float*)d_in[16];  // [VOCAB,DEC]
  const float* b_out      = (const float*)d_in[17];

  // outputs, concatenated flat: preds [B,30,VOCAB] | final_inds [B] | attn_w [B,30,S]
  float* preds      = (float*)d_out;
  float* out_finals = preds + (size_t)BATCH * NSTATES * VOCAB;
  float* out_attn   = out_finals + BATCH;

  // workspace bump allocator
  char* ws = (char*)d_ws;
  auto alloc_f = [&](size_t n) { float* p = (float*)ws; ws += n * sizeof(float); return p; };
  auto alloc_i = [&](size_t n) { int* p = (int*)ws;   ws += n * sizeof(int);   return p; };

  float* enc_proj = alloc_f((size_t)BATCH * SEQ * ATTN);   // [B,S,ATTN]
  float* states   = alloc_f((size_t)NSTATES * BATCH * DEC);
  float* emb      = alloc_f((size_t)BATCH * EMBD);
  float* gi_b     = alloc_f((size_t)BATCH * 3 * DEC);
  float* gh_b     = alloc_f((size_t)BATCH * 3 * DEC);
  float* s1       = alloc_f((size_t)BATCH * DEC);
  float* dec_proj = alloc_f((size_t)BATCH * ATTN);
  float* attn_vec = alloc_f((size_t)BATCH * ENC2);
  float* gi_t     = alloc_f((size_t)BATCH * 3 * ENC2);
  float* gh_t     = alloc_f((size_t)BATCH * 3 * ENC2);
  float* s2relu   = alloc_f((size_t)BATCH * ENC2);
  int*   inp      = alloc_i(BATCH);
  int*   stopped  = alloc_i(BATCH);
  int*   finals   = alloc_i(BATCH);

  auto gemm = [&](const float* A, const float* W, const float* bias, float* C,
                  int M, int N, int K, int bmode, int ldc) {
    dim3 grid(N / 64, M / 128);
    if (bmode == 0)
      gemm_wmma_f32<0><<<grid, 256, 0, stream>>>(A, W, bias, C, M, N, K, ldc);
    else
      gemm_wmma_f32<1><<<grid, 256, 0, stream>>>(A, W, bias, C, M, N, K, ldc);
  };

  init_state_kernel<<<1, BATCH, 0, stream>>>(inp, stopped, finals);

  // enc_proj = enc_states @ W_enc_a : [8192,1024] x [1024,256]
  gemm(enc_states, W_enc_a, nullptr, enc_proj, BATCH * SEQ, ATTN, ENC2, 1, ATTN);

  // one decode step: consume inp + prev state -> states[i], attn weights slice i
  auto process_step = [&](int i, const float* prev) {
    embed_kernel<<<BATCH, EMBD, 0, stream>>>(emb_table, inp, emb);
    gemm(emb,  Wih_b, bih_b, gi_b, BATCH, 3 * DEC, EMBD, 0, 3 * DEC);
    gemm(prev, Whh_b, bhh_b, gh_b, BATCH, 3 * DEC, DEC,  0, 3 * DEC);
    gru_combine_kernel<<<(BATCH * DEC + 255) / 256, 256, 0, stream>>>(
        gi_b, gh_b, prev, s1, DEC, 0);
    gemm(s1, W_dec_a, nullptr, dec_proj, BATCH, ATTN, DEC, 1, ATTN);
    attention_kernel<<<BATCH, 256, 0, stream>>>(
        enc_proj, dec_proj, v_a, enc_states,
        out_attn + (size_t)i * SEQ, NSTATES * SEQ, attn_vec);
    gemm(s1,       Wih_t, bih_t, gi_t, BATCH, 3 * ENC2, DEC,  0, 3 * ENC2);
    gemm(attn_vec, Whh_t, bhh_t, gh_t, BATCH, 3 * ENC2, ENC2, 0, 3 * ENC2);
    gru_combine_kernel<<<(BATCH * ENC2 + 255) / 256, 256, 0, stream>>>(
        gi_t, gh_t, attn_vec, s2relu, ENC2, 1);
    gemm(s2relu, W_fc, b_fc, states + (size_t)i * BATCH * DEC,
         BATCH, DEC, ENC2, 0, DEC);
  };

  // step 1: inp = BOS (set by init), prev = hidden -> states[0]
  process_step(0, hidden);

  // scan steps t = 2..30: logits(states[j]) -> preds[:,j,:], argmax -> states[j+1]
  for (int j = 0; j < NSTATES - 1; ++j) {
    const float* st = states + (size_t)j * BATCH * DEC;
    gemm(st, W_out, b_out, preds + (size_t)j * VOCAB,
         BATCH, VOCAB, DEC, 0, NSTATES * VOCAB);
    argmax_update_kernel<<<BATCH, 256, 0, stream>>>(
        preds + (size_t)j * VOCAB, NSTATES * VOCAB, inp, stopped, finals, 2 + j);
    process_step(j + 1, st);
  }
  // last state's logits go to preds only (never argmax'ed)
  gemm(states + (size_t)(NSTATES - 1) * BATCH * DEC, W_out, b_out,
       preds + (size_t)(NSTATES - 1) * VOCAB,
       BATCH, VOCAB, DEC, 0, NSTATES * VOCAB);

  finalize_kernel<<<1, BATCH, 0, stream>>>(finals, out_finals);
}